// FlowEmbedder_82798379532481
// MI455X (gfx1250) — compile-verified
//
#include <hip/hip_runtime.h>
#include <math.h>

// CDNA5 / gfx1250, wave32.
typedef __attribute__((ext_vector_type(2))) float v2f;
typedef __attribute__((ext_vector_type(8))) float v8f;

#define BLOCK 256
#define PPL   4            // pairs per lane
#define STEP 0.1f
#define FLOW_S 5.0f

// One lane handles PPL (i,j) pairs (strided so j / coalescing is preserved).
// Ragged t-loop, fp32 VALU MLP, v_exp_f32 normalizer, v_rsq_f32 for distance.
// Wave-level loss reduction via V_WMMA_F32_16X16X4_F32 (B = ones), halves
// combined with v_readlane, then LDS block reduction -> d_ws partial.
template<int N>
__global__ __launch_bounds__(BLOCK)
void flow_embed_loss_kernel(const float* __restrict__ pts,   // [N,2]
                            const float* __restrict__ W1, const float* __restrict__ b1,
                            const float* __restrict__ W2, const float* __restrict__ b2,
                            const float* __restrict__ W3, const float* __restrict__ b3,
                            const int*   __restrict__ gt,    // [N,N]
                            float* __restrict__ block_part,  // [gridDim.x]
                            int stride)                      // BLOCK * gridDim.x
{
    const int total = N * N;
    const int base  = blockIdx.x * BLOCK + threadIdx.x;

    // Uniform tiny-MLP weights (uniform addresses -> scalar loads).
    const float w100 = W1[0], w101 = W1[1], w110 = W1[2], w111 = W1[3];
    const float b10  = b1[0], b11  = b1[1];
    const float w200 = W2[0], w201 = W2[1], w210 = W2[2], w211 = W2[3];
    const float b20  = b2[0], b21  = b2[1];
    const float w300 = W3[0], w301 = W3[1], w310 = W3[2], w311 = W3[3];
    const float b30  = b3[0], b31  = b3[1];

    const float2* pts2 = (const float2*)pts;

    float pl = 0.0f;   // per-lane loss accumulator over PPL pairs

    #pragma unroll
    for (int k = 0; k < PPL; ++k) {
        const int flat  = base + k * stride;
        const bool valid = flat < total;
        const int f = valid ? flat : 0;
        const int i = f / N;          // N is a compile-time power of two -> shift
        const int j = f - i * N;

        const float2 p1 = pts2[i];
        const float2 p2 = pts2[j];    // coalesced in j

        const float dx = p2.x - p1.x;
        const float dy = p2.y - p1.y;
        const float d2 = fmaf(dx, dx, dy * dy);
        const float r  = __builtin_amdgcn_rsqf(d2);       // v_rsq_f32
        const bool  zero = (d2 < 1e-16f);                 // euc < 1e-8
        const float euc = zero ? 0.0f : d2 * r;
        const float inv = zero ? STEP : STEP * r;         // any finite value ok when diff==0
        const float sx = dx * inv;                        // per-step displacement
        const float sy = dy * inv;

        int tnum = (int)(euc * 10.0f) + 1;                // t <= floor(euc/STEP)
        if (tnum > 16) tnum = 16;

        float D = 0.0f;
        for (int t = 0; t < tnum; ++t) {
            const float tf = (float)t;
            const float px = fmaf(tf, sx, p1.x);
            const float py = fmaf(tf, sy, p1.y);
            // layer 1 (y = x @ W^T + b), ReLU
            const float h10 = fmaxf(fmaf(w100, px, fmaf(w101, py, b10)), 0.0f);
            const float h11 = fmaxf(fmaf(w110, px, fmaf(w111, py, b11)), 0.0f);
            // layer 2, ReLU
            const float h20 = fmaxf(fmaf(w200, h10, fmaf(w201, h11, b20)), 0.0f);
            const float h21 = fmaxf(fmaf(w210, h10, fmaf(w211, h11, b21)), 0.0f);
            // layer 3
            const float fx = fmaf(w300, h20, fmaf(w301, h21, b30));
            const float fy = fmaf(w310, h20, fmaf(w311, h21, b31));
            const float dot = fmaf(fx, dx, fy * dy);
            // STEP/(S/(1+(S-1)e^-d)) = STEP/S + STEP*(S-1)/S * e^-d
            D += fmaf((STEP * (FLOW_S - 1.0f) / FLOW_S), __expf(-dot), (STEP / FLOW_S));
        }
        D = zero ? 0.0f : D;

        // D >= 0.02 here unless zeroed, gt in [0,9]: plain __logf(1+x) is safe.
        const float gl = __logf((float)(gt[f] + 1));
        const float dl = __logf(1.0f + D);
        const float e  = gl - dl;
        pl += valid ? e * e : 0.0f;
    }

    // ---- wave32 reduction via WMMA: D = A(16x4) * ones(4x16) + 0 ----
    // A VGPR0 = one partial per lane (K=0 rows 0..15 from lanes 0..15, K=2
    // from lanes 16..31); A VGPR1 = 0. Summing the 8 accumulator VGPRs gives
    // lanes 0..15: sum(rows 0..7), lanes 16..31: sum(rows 8..15).
    v2f a;  a.x = pl;    a.y = 0.0f;
    v2f bb; bb.x = 1.0f; bb.y = 1.0f;
    v8f c = {};
    c = __builtin_amdgcn_wmma_f32_16x16x4_f32(
            /*neg_a=*/false, a, /*neg_b=*/false, bb,
            /*c_mod=*/(short)0, c, /*reuse_a=*/false, /*reuse_b=*/false);
    float s = c[0] + c[1] + c[2] + c[3] + c[4] + c[5] + c[6] + c[7];

    // Combine the two half-wave row-sums with readlanes (SGPR result).
    const int si = __builtin_bit_cast(int, s);
    const float tot = __builtin_bit_cast(float, __builtin_amdgcn_readlane(si, 0)) +
                      __builtin_bit_cast(float, __builtin_amdgcn_readlane(si, 16));

    // ---- block reduction over the 8 waves ----
    __shared__ float lds[BLOCK / 32];
    const int lane = threadIdx.x & 31;
    const int wave = threadIdx.x >> 5;
    if (lane == 0) lds[wave] = tot;
    __syncthreads();
    if (threadIdx.x == 0) {
        float bs = 0.0f;
        #pragma unroll
        for (int w = 0; w < BLOCK / 32; ++w) bs += lds[w];
        block_part[blockIdx.x] = bs;
    }
}

// Deterministic fixed-order final reduction (single block).
__global__ __launch_bounds__(BLOCK)
void flow_embed_final_reduce(const float* __restrict__ block_part, int nblocks,
                             float* __restrict__ out)
{
    __shared__ float lds[BLOCK];
    float s = 0.0f;
    for (int k = threadIdx.x; k < nblocks; k += BLOCK) s += block_part[k];
    lds[threadIdx.x] = s;
    __syncthreads();
    for (int off = BLOCK / 2; off > 0; off >>= 1) {
        if ((int)threadIdx.x < off) lds[threadIdx.x] += lds[threadIdx.x + off];
        __syncthreads();
    }
    if (threadIdx.x == 0) out[0] = lds[0];
}

extern "C" void kernel_launch(void* const* d_in, const int* in_sizes, int n_in,
                              void* d_out, int out_size, void* d_ws, size_t ws_size,
                              hipStream_t stream) {
    (void)n_in; (void)out_size; (void)ws_size;
    const float* pts = (const float*)d_in[0];
    const float* W1  = (const float*)d_in[1];
    const float* b1  = (const float*)d_in[2];
    const float* W2  = (const float*)d_in[3];
    const float* b2  = (const float*)d_in[4];
    const float* W3  = (const float*)d_in[5];
    const float* b3  = (const float*)d_in[6];
    const int*   gt  = (const int*)d_in[7];
    float* out  = (float*)d_out;
    float* part = (float*)d_ws;

    constexpr int N = 1024;                         // reference fixes N
    const int total   = N * N;                      // 1,048,576
    const int nblocks = total / (BLOCK * PPL);      // 1024
    const int stride  = BLOCK * nblocks;            // 262,144 (multiple of N)

    flow_embed_loss_kernel<N><<<nblocks, BLOCK, 0, stream>>>(
        pts, W1, b1, W2, b2, W3, b3, gt, part, stride);
    flow_embed_final_reduce<<<1, BLOCK, 0, stream>>>(part, nblocks, out);
}